// Generator3DLUT_identity_73057393705704
// MI455X (gfx1250) — compile-verified
//
#include <hip/hip_runtime.h>

// Problem constants (fixed by the reference harness).
#define BATCH 8
#define IMG_H 1080
#define IMG_W 1920
#define HW    (IMG_H * IMG_W)          // 2,073,600 (divisible by 4)
#define LDIM  33
#define LDIM2 (LDIM * LDIM)
#define NLUT  (LDIM * LDIM * LDIM)     // 35937

// (dim-1)/1.0001 : reciprocal of binsize, precomputed in double then rounded.
#define INV_BINSIZE 31.996800319968003f

typedef float v4f __attribute__((ext_vector_type(4)));

// Elementwise lerp; -ffp-contract fuses into v_fma. The .w lane is dead and DCE'd.
__device__ __forceinline__ v4f lerp4(const v4f a, const v4f b, const float t) {
    return a + t * (b - a);
}

// Repack planar LUT (3, 35937) -> AoS v4f so one B128 gather fetches all 3 channels.
// Regular-temporal store: we *want* the packed LUT resident in L2.
__global__ __launch_bounds__(256) void lut_repack_kernel(const float* __restrict__ lut,
                                                         v4f* __restrict__ lut4) {
    int i = blockIdx.x * blockDim.x + threadIdx.x;
    if (i < NLUT) {
        v4f v;
        v.x = lut[i];
        v.y = lut[NLUT + i];
        v.z = lut[2 * NLUT + i];
        v.w = 0.0f;
        lut4[i] = v;
    }
}

// Each thread handles 4 consecutive pixels of one image:
//   3x B128 NT plane loads, 8x B128 RT LUT gathers per pixel, 3x B128 NT plane stores.
// NT hints keep the 2x199MB streams from evicting the 575KB LUT out of L2.
template <bool PACKED>
__global__ __launch_bounds__(256) void lut3d_apply_kernel(const float* __restrict__ x,
                                                          const float* __restrict__ lut,  // planar fallback
                                                          const v4f* __restrict__ lut4,   // packed path
                                                          float* __restrict__ out,
                                                          int n4) {
    int t = blockIdx.x * blockDim.x + threadIdx.x;
    if (t >= n4) return;

    const int p   = t << 2;          // first pixel of this thread's group of 4
    const int img = p / HW;          // batch index
    const int rem = p - img * HW;    // position within plane
    const int off = img * 3 * HW + rem;

    const v4f r4 = __builtin_nontemporal_load(reinterpret_cast<const v4f*>(x + off));
    const v4f g4 = __builtin_nontemporal_load(reinterpret_cast<const v4f*>(x + off + HW));
    const v4f b4 = __builtin_nontemporal_load(reinterpret_cast<const v4f*>(x + off + 2 * HW));

    // gfx1250 global_prefetch_b8: pull the stream ahead; speculative, drop-on-fault.
    __builtin_prefetch(x + off + 4096, 0, 1);
    __builtin_prefetch(x + off + HW + 4096, 0, 1);
    __builtin_prefetch(x + off + 2 * HW + 4096, 0, 1);

    const float rr[4] = {r4.x, r4.y, r4.z, r4.w};
    const float gg[4] = {g4.x, g4.y, g4.z, g4.w};
    const float bb[4] = {b4.x, b4.y, b4.z, b4.w};
    float o0[4], o1[4], o2[4];

#pragma unroll
    for (int k = 0; k < 4; ++k) {
        const float rf = rr[k] * INV_BINSIZE;
        const float gf = gg[k] * INV_BINSIZE;
        const float bf = bb[k] * INV_BINSIZE;

        // inputs are in [0,1): floor == trunc; clamp to dim-2 = 31.
        int rid = (int)rf; rid = rid < 31 ? rid : 31;
        int gid = (int)gf; gid = gid < 31 ? gid : 31;
        int bid = (int)bf; bid = bid < 31 ? bid : 31;

        const float fr = rf - (float)rid;
        const float fg = gf - (float)gid;
        const float fb = bf - (float)bid;

        const int base = bid * LDIM2 + gid * LDIM + rid;

        v4f c000, c001, c010, c011, c100, c101, c110, c111;
        if (PACKED) {
            c000 = lut4[base];
            c001 = lut4[base + 1];
            c010 = lut4[base + LDIM];
            c011 = lut4[base + LDIM + 1];
            c100 = lut4[base + LDIM2];
            c101 = lut4[base + LDIM2 + 1];
            c110 = lut4[base + LDIM2 + LDIM];
            c111 = lut4[base + LDIM2 + LDIM + 1];
        } else {
            const float* l0 = lut;
            const float* l1 = lut + NLUT;
            const float* l2 = lut + 2 * NLUT;
            const int i000 = base,                i001 = base + 1;
            const int i010 = base + LDIM,         i011 = base + LDIM + 1;
            const int i100 = base + LDIM2,        i101 = base + LDIM2 + 1;
            const int i110 = base + LDIM2 + LDIM, i111 = base + LDIM2 + LDIM + 1;
            c000 = (v4f){l0[i000], l1[i000], l2[i000], 0.f};
            c001 = (v4f){l0[i001], l1[i001], l2[i001], 0.f};
            c010 = (v4f){l0[i010], l1[i010], l2[i010], 0.f};
            c011 = (v4f){l0[i011], l1[i011], l2[i011], 0.f};
            c100 = (v4f){l0[i100], l1[i100], l2[i100], 0.f};
            c101 = (v4f){l0[i101], l1[i101], l2[i101], 0.f};
            c110 = (v4f){l0[i110], l1[i110], l2[i110], 0.f};
            c111 = (v4f){l0[i111], l1[i111], l2[i111], 0.f};
        }

        const v4f cA = lerp4(lerp4(c000, c001, fr), lerp4(c010, c011, fr), fg);
        const v4f cB = lerp4(lerp4(c100, c101, fr), lerp4(c110, c111, fr), fg);
        const v4f c  = lerp4(cA, cB, fb);

        o0[k] = c.x;
        o1[k] = c.y;
        o2[k] = c.z;
    }

    const v4f outR = {o0[0], o0[1], o0[2], o0[3]};
    const v4f outG = {o1[0], o1[1], o1[2], o1[3]};
    const v4f outB = {o2[0], o2[1], o2[2], o2[3]};
    __builtin_nontemporal_store(outR, reinterpret_cast<v4f*>(out + off));
    __builtin_nontemporal_store(outG, reinterpret_cast<v4f*>(out + off + HW));
    __builtin_nontemporal_store(outB, reinterpret_cast<v4f*>(out + off + 2 * HW));
}

extern "C" void kernel_launch(void* const* d_in, const int* in_sizes, int n_in,
                              void* d_out, int out_size, void* d_ws, size_t ws_size,
                              hipStream_t stream) {
    const float* lut = (const float*)d_in[0]; // (3, 33, 33, 33)
    const float* x   = (const float*)d_in[1]; // (8, 3, 1080, 1920)
    float* out       = (float*)d_out;         // (8, 3, 1080, 1920)

    const int npix = out_size / 3;   // B*H*W
    const int n4   = npix >> 2;      // threads (4 pixels each)

    const size_t need = (size_t)NLUT * sizeof(v4f);
    const int threads = 256;

    if (ws_size >= need) {
        v4f* lut4 = (v4f*)d_ws;
        lut_repack_kernel<<<(NLUT + threads - 1) / threads, threads, 0, stream>>>(lut, lut4);
        lut3d_apply_kernel<true><<<(n4 + threads - 1) / threads, threads, 0, stream>>>(
            x, lut, lut4, out, n4);
    } else {
        lut3d_apply_kernel<false><<<(n4 + threads - 1) / threads, threads, 0, stream>>>(
            x, lut, nullptr, out, n4);
    }
}